// Model_39676907883207
// MI455X (gfx1250) — compile-verified
//
#include <hip/hip_runtime.h>

// out = (Q K^T / 8) V  ==  Q (K^T V) / 8   (softmax in reference is dead code)
// B=8, S=2048, D=128.
// Phase 1: M[b] = K[b]^T V[b] (128x128 per batch, 512 KB total in d_ws),
//          LDS-staged, WMMA f32 16x16x4, atomic FP32 reduction over S-chunks.
// Phase 2: out[b] = Q[b] M[b] * 0.125, M column-panel staged in LDS per WG.

typedef __attribute__((ext_vector_type(2))) float v2f;
typedef __attribute__((ext_vector_type(8))) float v8f;

#define B_   8
#define S_   2048
#define D_   128
#define NS_  8      // S-chunks per M-tile in phase 1
#define TT_  32     // rows staged per LDS block in phase 1
#define VST_ 136    // padded LDS row stride (floats): 2*136 mod 64 banks = 16 -> halves disjoint

// ---------------------------------------------------------------------------
// Phase 1: M[b] = K[b]^T @ V[b].
// Workgroup: 256 threads = 8 waves; wave w computes tile (ti, tj=w).
// Grid: B * 8(ti) * NS_ workgroups.
// ---------------------------------------------------------------------------
__global__ __launch_bounds__(256) void ktv_kernel(const float* __restrict__ K,
                                                  const float* __restrict__ V,
                                                  float* __restrict__ M) {
  __shared__ float Kl[TT_ * 16];    // K panel: TT_ rows x 16 cols (cols of tile ti)
  __shared__ float Vl[TT_ * VST_];  // V block: TT_ rows x 128 cols, padded stride

  int idx = blockIdx.x;
  const int chunk = idx % NS_;       idx /= NS_;
  const int ti    = idx % (D_ / 16);
  const int b     = idx / (D_ / 16);

  const int tid  = threadIdx.x;
  const int tj   = tid >> 5;          // wave id = output tile column
  const int lane = tid & 31;
  const int m    = lane & 15;
  const int koff = (lane >> 4) << 1;  // lanes 0-15 -> K={0,1}; 16-31 -> K={2,3}

  const int t0 = chunk * (S_ / NS_);

  // ---- per-thread staging bases (constant across block iterations)
  const int vrow = tid >> 5;           // 0..7 (rows advance by 8 per unroll step)
  const int vc4  = (tid & 31) << 2;    // 0..124 step 4
  const float* __restrict__ vsrc =
      V + ((size_t)b * S_ + t0 + vrow) * D_ + vc4;
  float* const vdst = &Vl[vrow * VST_ + vc4];

  const int krow = tid >> 2;           // 0..63 (only 0..31 used)
  const int kc4  = (tid & 3) << 2;     // 0..12 step 4
  const float* __restrict__ ksrc =
      K + ((size_t)b * S_ + t0 + krow) * D_ + ti * 16 + kc4;
  float* const kdst = &Kl[krow * 16 + kc4];
  const bool kstager = (tid < TT_ * 16 / 4);   // threads 0-127

  // ---- per-lane fragment bases (immediate-offset DS loads in the hot loop)
  const float* const afrag = &Kl[koff * 16 + m];
  const float* const bfrag = &Vl[koff * VST_ + tj * 16 + m];

  v8f c = {};

  for (int tb = 0; tb < S_ / NS_; tb += TT_) {
    // ---- load pass: 5 global_load_b128 in flight (distinct dest regs)
    float4 kt = {};
    if (kstager) kt = *(const float4*)ksrc;
    float4 vt[4];
#pragma unroll
    for (int j = 0; j < 4; ++j) {
      __builtin_prefetch(vsrc + (size_t)(TT_ + j * 8) * D_, 0, 1);
      vt[j] = *(const float4*)(vsrc + (size_t)j * 8 * D_);
    }
    // ---- store pass: drain into LDS (compiler uses partial loadcnt waits)
#pragma unroll
    for (int j = 0; j < 4; ++j) {
      *(float4*)(vdst + j * 8 * VST_) = vt[j];
    }
    if (kstager) *(float4*)kdst = kt;
    __syncthreads();

    // ---- 8 WMMAs per block, operands via ds_load_2addr at constant offsets
#pragma unroll
    for (int tt = 0; tt < TT_; tt += 4) {
      v2f a, bb;
      a.x  = afrag[(tt)     * 16];
      a.y  = afrag[(tt + 1) * 16];
      bb.x = bfrag[(tt)     * VST_];
      bb.y = bfrag[(tt + 1) * VST_];
      c = __builtin_amdgcn_wmma_f32_16x16x4_f32(false, a, false, bb,
                                                (short)0, c, false, false);
    }
    __syncthreads();

    vsrc += (size_t)TT_ * D_;
    ksrc += (size_t)TT_ * D_;
  }

  // C/D layout: VGPR v -> row v (lanes 0-15) / v+8 (lanes 16-31); col = lane&15
  float* __restrict__ Mb = M + (size_t)b * D_ * D_;
  const int rbase = ti * 16 + (lane >> 4) * 8;
  const int col   = tj * 16 + (lane & 15);
#pragma unroll
  for (int v = 0; v < 8; ++v) {
    __hip_atomic_fetch_add(&Mb[(size_t)(rbase + v) * D_ + col], c[v],
                           __ATOMIC_RELAXED, __HIP_MEMORY_SCOPE_AGENT);
  }
}

// ---------------------------------------------------------------------------
// Phase 2: out[b] = Q[b] @ M[b] * 0.125.
// Workgroup: 256 threads = 8 waves = 8 consecutive si tiles, one tj panel.
// M column-panel (128x16 = 8 KB) staged in LDS once per workgroup.
// Grid: B * (S/16/8) * 8(tj) workgroups.
// ---------------------------------------------------------------------------
__global__ __launch_bounds__(256) void qm_kernel(const float* __restrict__ Q,
                                                 const float* __restrict__ M,
                                                 float* __restrict__ O) {
  __shared__ float Ml[D_ * 16];  // 128 rows x 16 cols, stride 16 (conflict-free)

  int idx = blockIdx.x;
  const int tj = idx % (D_ / 16);       idx /= (D_ / 16);
  const int sg = idx % (S_ / 16 / 8);
  const int b  = idx / (S_ / 16 / 8);

  const int tid  = threadIdx.x;
  const int wave = tid >> 5;
  const int lane = tid & 31;
  const int m    = lane & 15;
  const int koff = (lane >> 4) << 1;

  // ---- stage M panel: load both b128 first, then store both
  const int mrow = tid >> 2;           // 0..63 (rows advance by 64 per step)
  const int mc4  = (tid & 3) << 2;
  const float* __restrict__ msrc =
      M + (size_t)b * D_ * D_ + (size_t)mrow * D_ + tj * 16 + mc4;
  float* const mdst = &Ml[mrow * 16 + mc4];
  float4 mt[2];
#pragma unroll
  for (int j = 0; j < 2; ++j) {
    mt[j] = *(const float4*)(msrc + (size_t)j * 64 * D_);
  }
#pragma unroll
  for (int j = 0; j < 2; ++j) {
    *(float4*)(mdst + j * 64 * 16) = mt[j];
  }
  __syncthreads();

  const int si = sg * 8 + wave;
  const float* __restrict__ Qrow = Q + ((size_t)b * S_ + si * 16 + m) * D_;
  const float* const bfrag = &Ml[koff * 16 + m];

  v8f c = {};
#pragma unroll
  for (int k0 = 0; k0 < D_; k0 += 4) {
    v2f a = *(const v2f*)(Qrow + k0 + koff);   // contiguous 8B-aligned -> b64
    v2f bb;
    bb.x = bfrag[(k0)     * 16];
    bb.y = bfrag[(k0 + 1) * 16];
    c = __builtin_amdgcn_wmma_f32_16x16x4_f32(false, a, false, bb,
                                              (short)0, c, false, false);
  }

  float* __restrict__ Ob = O + ((size_t)b * S_ + si * 16) * D_;
  const int rbase = (lane >> 4) * 8;
  const int col   = tj * 16 + (lane & 15);
#pragma unroll
  for (int v = 0; v < 8; ++v) {
    Ob[(size_t)(rbase + v) * D_ + col] = c[v] * 0.125f;
  }
}

// ---------------------------------------------------------------------------
extern "C" void kernel_launch(void* const* d_in, const int* in_sizes, int n_in,
                              void* d_out, int out_size, void* d_ws, size_t ws_size,
                              hipStream_t stream) {
  const float* q = (const float*)d_in[0];
  const float* k = (const float*)d_in[1];
  const float* v = (const float*)d_in[2];
  float*       o = (float*)d_out;
  float*       M = (float*)d_ws;   // B * D * D floats = 512 KB

  (void)in_sizes; (void)n_in; (void)out_size; (void)ws_size;

  hipMemsetAsync(M, 0, (size_t)B_ * D_ * D_ * sizeof(float), stream);

  ktv_kernel<<<B_ * (D_ / 16) * NS_,           256, 0, stream>>>(k, v, M);
  qm_kernel <<<B_ * (S_ / 16 / 8) * (D_ / 16), 256, 0, stream>>>(q, M, o);
}